// ScaledDotProductAttention_14207751815439
// MI455X (gfx1250) — compile-verified
//
#include <hip/hip_runtime.h>

// CDNA5 (gfx1250) fused bilinear attention:
//   S = scale * (Q @ W) @ K^T ; P = softmax(S) ; out = P @ V ; also emit P.
// fp32 WMMA (V_WMMA_F32_16X16X4_F32) throughout — problem is HBM-bound on the
// 537 MB attn_weights write, so full fp32 precision is free.

typedef float v2f __attribute__((ext_vector_type(2)));
typedef float v8f __attribute__((ext_vector_type(8)));

#define BATCH   2
#define HEADS   16
#define L_SEQ   2048
#define D_HEAD  64
#define ROWS    16          // Q rows per workgroup
#define SROW    2052        // padded LDS row stride for score tile (2052 % 64 == 4)
#define QROW    68          // padded LDS row stride for qw tile   (68 % 64 == 4)
#define NWAVES  8           // 256 threads = 8 wave32

__global__ __launch_bounds__(256)
void bilinear_attn_fused(const float* __restrict__ q,
                         const float* __restrict__ k,
                         const float* __restrict__ v,
                         const float* __restrict__ w,
                         float* __restrict__ out,    // [B*H, L, D]
                         float* __restrict__ attn)   // [B*H, L, L]
{
    extern __shared__ float smem[];
    float* Stile = smem;                      // ROWS * SROW  (score / P tile)
    float* qw    = Stile + ROWS * SROW;       // ROWS * QROW  (Q @ W tile)
    float* red   = qw + ROWS * QROW;          // 4 * 256     (phase-4 reduction)

    const int bh   = blockIdx.y;
    const int r0   = blockIdx.x * ROWS;
    const int tid  = threadIdx.x;
    const int lane = tid & 31;
    const int wave = tid >> 5;
    const int l16  = lane & 15;          // M (or N) index within fragment
    const int lh2  = (lane >> 4) << 1;   // K sub-offset: 0 for lanes 0-15, 2 for 16-31
    const int rowh = (lane >> 4) << 3;   // C/D row offset: 0 or 8

    const float scale = rsqrtf((float)D_HEAD);

    const float* Qb = q + ((size_t)bh * L_SEQ + r0) * D_HEAD;
    const float* Kb = k + (size_t)bh * L_SEQ * D_HEAD;
    const float* Vb = v + (size_t)bh * L_SEQ * D_HEAD;
    const float* Wb = w + (size_t)bh * D_HEAD * D_HEAD;

    // ---------------- Phase 1: qw = Qtile(16x64) @ W(64x64), waves 0-3 -------
    if (wave < 4) {
        const int n0 = wave * 16;
        v8f acc = {};
        for (int kkb = 0; kkb < D_HEAD; kkb += 4) {
            const int kb = kkb + lh2;
            v2f a = *(const v2f*)(Qb + (size_t)l16 * D_HEAD + kb);   // A: Q rows
            v2f b;                                                   // B: W cols
            b.x = Wb[(size_t)kb * D_HEAD + n0 + l16];
            b.y = Wb[(size_t)(kb + 1) * D_HEAD + n0 + l16];
            acc = __builtin_amdgcn_wmma_f32_16x16x4_f32(false, a, false, b,
                                                        (short)0, acc, false, false);
        }
#pragma unroll
        for (int i = 0; i < 8; ++i)
            qw[(i + rowh) * QROW + n0 + l16] = acc[i];
    }
    __syncthreads();

    // ---------------- Phase 2: Stile = scale * qw @ K^T  (all 8 waves) -------
    // Hoist the 16 A-fragments (whole 16x64 qw tile) into VGPRs once.
    v2f afr[16];
#pragma unroll
    for (int t = 0; t < 16; ++t) {
        const int kb = 4 * t + lh2;
        afr[t].x = qw[l16 * QROW + kb];
        afr[t].y = qw[l16 * QROW + kb + 1];
    }
    for (int blk = wave; blk < L_SEQ / 16; blk += NWAVES) {
        const int n0 = blk * 16;
        v8f acc = {};
        // B = K^T: B[k'][n] = Keys[n][k'] -> contiguous 8B loads per lane
        const float* Kc = Kb + (size_t)(n0 + l16) * D_HEAD + lh2;
#pragma unroll
        for (int t = 0; t < 16; ++t) {
            v2f b = *(const v2f*)(Kc + 4 * t);
            acc = __builtin_amdgcn_wmma_f32_16x16x4_f32(false, afr[t], false, b,
                                                        (short)0, acc, false, false);
        }
#pragma unroll
        for (int i = 0; i < 8; ++i)
            Stile[(i + rowh) * SROW + n0 + l16] = acc[i] * scale;
    }
    __syncthreads();

    // ---------------- Phase 3: row softmax in LDS; stream P to global --------
    for (int r = wave; r < ROWS; r += NWAVES) {
        float* Srow = Stile + r * SROW;
        float m = -3.402823466e38f;
        for (int i = lane; i < L_SEQ; i += 32) m = fmaxf(m, Srow[i]);
#pragma unroll
        for (int off = 16; off > 0; off >>= 1) m = fmaxf(m, __shfl_xor(m, off, 32));
        float s = 0.f;
        for (int i = lane; i < L_SEQ; i += 32) {
            float e = __expf(Srow[i] - m);
            Srow[i] = e;
            s += e;
        }
#pragma unroll
        for (int off = 16; off > 0; off >>= 1) s += __shfl_xor(s, off, 32);
        const float inv = 1.0f / s;
        float* Arow = attn + ((size_t)bh * L_SEQ + (r0 + r)) * L_SEQ;
        for (int i = lane; i < L_SEQ; i += 32) {   // coalesced 537MB write (the roofline)
            const float p = Srow[i] * inv;
            Srow[i] = p;
            Arow[i] = p;
        }
    }
    __syncthreads();

    // ---------------- Phase 4: out = P(16x2048) @ V(2048x64) ----------------
    // wave -> (ntile = wave&3, K-half = wave>>2); pairs reduce via LDS.
    {
        const int n0   = (wave & 3) * 16;
        const int kbeg = (wave >> 2) * (L_SEQ / 2);
        v8f acc = {};
        for (int kkb = kbeg; kkb < kbeg + L_SEQ / 2; kkb += 4) {
            const int kb = kkb + lh2;
            v2f a;                                    // A = P rows from LDS
            a.x = Stile[l16 * SROW + kb];
            a.y = Stile[l16 * SROW + kb + 1];
            v2f b;                                    // B = V
            b.x = Vb[(size_t)kb * D_HEAD + n0 + l16];
            b.y = Vb[(size_t)(kb + 1) * D_HEAD + n0 + l16];
            acc = __builtin_amdgcn_wmma_f32_16x16x4_f32(false, a, false, b,
                                                        (short)0, acc, false, false);
        }
        if (wave >= 4) {
#pragma unroll
            for (int i = 0; i < 8; ++i)
                red[(wave - 4) * 256 + i * 32 + lane] = acc[i];
        }
        __syncthreads();
        if (wave < 4) {
#pragma unroll
            for (int i = 0; i < 8; ++i)
                acc[i] += red[wave * 256 + i * 32 + lane];
            float* Ob = out + ((size_t)bh * L_SEQ + r0) * D_HEAD;
#pragma unroll
            for (int i = 0; i < 8; ++i)
                Ob[(size_t)(i + rowh) * D_HEAD + n0 + l16] = acc[i];
        }
    }
}

extern "C" void kernel_launch(void* const* d_in, const int* in_sizes, int n_in,
                              void* d_out, int out_size, void* d_ws, size_t ws_size,
                              hipStream_t stream) {
    (void)in_sizes; (void)n_in; (void)out_size; (void)d_ws; (void)ws_size;
    const float* q = (const float*)d_in[0];   // [B,H,L,D]
    const float* k = (const float*)d_in[1];   // [B,H,L,D]
    const float* v = (const float*)d_in[2];   // [B,H,L,D]
    const float* w = (const float*)d_in[3];   // [B,H,D,D]
    float* out  = (float*)d_out;                                      // [B,H,L,D]
    float* attn = out + (size_t)BATCH * HEADS * L_SEQ * D_HEAD;       // [B,H,L,L]

    const size_t smem = (size_t)(ROWS * SROW + ROWS * QROW + 4 * 256) * sizeof(float);
    static bool attr_set = false;   // idempotent, deterministic
    if (!attr_set) {
        hipFuncSetAttribute((const void*)bilinear_attn_fused,
                            hipFuncAttributeMaxDynamicSharedMemorySize, (int)smem);
        attr_set = true;
    }
    dim3 grid(L_SEQ / ROWS, BATCH * HEADS);   // 128 x 32 workgroups
    bilinear_attn_fused<<<grid, 256, smem, stream>>>(q, k, v, w, out, attn);
}